// Node2Point_35064113005042
// MI455X (gfx1250) — compile-verified
//
#include <hip/hip_runtime.h>
#include <hip/hip_bf16.h>
#include <math.h>

// dims fixed by the reference
#define NB_  16
#define K_   64
#define L_   1024   // NB_*K_
#define M_   450    // FINEMATCH_MAX_POINT
#define D_   256
#define SLICES_ 8   // row-slices per proposal in the copy kernel
#define G_   4      // rows batched per wave per async group

typedef unsigned long long ull_t;

// -------- CDNA5 async LDS<->global DMA helpers (ASYNCcnt-tracked) ----------
__device__ __forceinline__ void async_wait0() {
    asm volatile("s_wait_asynccnt 0x0" ::: "memory");
}
__device__ __forceinline__ void async_load_b128(unsigned lds_off, ull_t gaddr) {
    asm volatile("global_load_async_to_lds_b128 %0, %1, off"
                 :: "v"(lds_off), "v"(gaddr) : "memory");
}
__device__ __forceinline__ void async_load_b128_o512(unsigned lds_off, ull_t gaddr) {
    asm volatile("global_load_async_to_lds_b128 %0, %1, off offset:512"
                 :: "v"(lds_off), "v"(gaddr) : "memory");
}
__device__ __forceinline__ void async_store_b128(ull_t gaddr, unsigned lds_off) {
    asm volatile("global_store_async_from_lds_b128 %0, %1, off"
                 :: "v"(gaddr), "v"(lds_off) : "memory");
}
__device__ __forceinline__ void async_store_b128_o512(ull_t gaddr, unsigned lds_off) {
    asm volatile("global_store_async_from_lds_b128 %0, %1, off offset:512"
                 :: "v"(gaddr), "v"(lds_off) : "memory");
}
// generic pointer to shared memory: low 32 bits are the LDS byte offset
__device__ __forceinline__ unsigned lds_off_of(const void* p) {
    return (unsigned)(ull_t)(uintptr_t)p;
}

// ===========================================================================
// Kernel 1: per-proposal stable compaction -> workspace
//   ws_n[p], ws_fidx[p][1024], ws_pts[p][1024][3]
// ===========================================================================
__global__ __launch_bounds__(256)
void n2p_compact_kernel(const int*   __restrict__ nb_mask,
                        const int*   __restrict__ seed_idx,
                        const int*   __restrict__ knn_mask,
                        const float* __restrict__ knn_pts,
                        const int*   __restrict__ knn_idx,
                        int* __restrict__ ws_n,
                        int* __restrict__ ws_fidx,
                        float* __restrict__ ws_pts) {
    __shared__ int sh_scan[256];
    const int p = blockIdx.x;
    const int t = threadIdx.x;

    const int s0   = t * 4;
    const int nb   = s0 >> 6;
    const int kk   = s0 & 63;
    const int node = seed_idx[p * NB_ + nb];
    const bool nbm = (nb_mask[p * NB_ + nb] != 0);

    int  fidx[4];
    bool vld [4];
    int  cnt = 0;
#pragma unroll
    for (int j = 0; j < 4; ++j) {
        const int slot = node * K_ + kk + j;
        vld[j]  = nbm && (knn_mask[slot] != 0);
        fidx[j] = knn_idx[slot];
        cnt += vld[j] ? 1 : 0;
    }
    sh_scan[t] = cnt;
    __syncthreads();
#pragma unroll
    for (int off = 1; off < 256; off <<= 1) {
        int add = 0;
        if (t >= off) add = sh_scan[t - off];
        __syncthreads();
        sh_scan[t] += add;
        __syncthreads();
    }
    int pos = sh_scan[t] - cnt;              // stable exclusive prefix
    int*   fdst = ws_fidx + p * L_;
    float* pdst = ws_pts  + (size_t)p * L_ * 3;
#pragma unroll
    for (int j = 0; j < 4; ++j) {
        if (vld[j]) {
            fdst[pos] = fidx[j];
            const float* pp = knn_pts + (size_t)(node * K_ + kk + j) * 3;
            pdst[pos * 3 + 0] = pp[0];
            pdst[pos * 3 + 1] = pp[1];
            pdst[pos * 3 + 2] = pp[2];
            ++pos;
        }
    }
    if (t == 255) ws_n[p] = sh_scan[255];
}

// ===========================================================================
// Kernel 2: resample + feature-row gather via batched async LDS DMA
//   grid (P, SLICES_), 256 threads = 8 wave32; each wave batches G_ rows
// ===========================================================================
__global__ __launch_bounds__(256)
void n2p_copy_kernel(const int*   __restrict__ ws_n,
                     const int*   __restrict__ ws_fidx,
                     const float* __restrict__ ws_pts,
                     const float* __restrict__ feats,
                     float* __restrict__ out_pts,
                     float* __restrict__ out_feats,
                     float* __restrict__ out_mask,
                     int npts) {
    __shared__ __align__(16) float sh_stage[8 * G_ * D_];   // 8 waves x 4KB

    const int p     = blockIdx.x;
    const int slice = blockIdx.y;
    const int wave  = threadIdx.x >> 5;
    const int lane  = threadIdx.x & 31;

    const int rows_per_slice = (M_ + SLICES_ - 1) / SLICES_;   // 57
    const int i_begin = slice * rows_per_slice;
    const int i_end   = (i_begin + rows_per_slice < M_) ? (i_begin + rows_per_slice) : M_;

    const int   n        = ws_n[p];          // uniform
    const int   n_eff    = (n < M_) ? n : M_;
    const bool  resample = (n > M_);
    const float nf       = (float)n;

    const int*   fbase = ws_fidx + p * L_;
    const float* pbase = ws_pts  + (size_t)p * L_ * 3;
    float* outp = out_pts  + (size_t)p * M_ * 3;
    float* outf = out_feats + (size_t)p * M_ * D_;
    float* outm = out_mask + (size_t)p * M_;

    // this wave's staging base (+lane*16B); slot g at +g*1024 bytes
    const unsigned stage = lds_off_of(&sh_stage[wave * G_ * D_ + lane * 4]);

    for (int ib = i_begin + wave; ib < i_end; ib += 8 * G_) {
        int rfi[G_];
        // reclaim all of this wave's slots (previous group's stores done)
        async_wait0();
        // ---- pass 1: bookkeeping, pts/mask, zero rows, issue async loads ----
#pragma unroll
        for (int g = 0; g < G_; ++g) {
            const int i = ib + 8 * g;
            rfi[g] = -1;
            if (i >= i_end) continue;
            float* dstf = outf + (size_t)i * D_;
            if (i < n_eff) {                      // wave-uniform condition
                int sel = resample ? (int)floorf(((float)i / 450.0f) * nf) : i;
                if (sel > L_ - 1) sel = L_ - 1;   // mirrors reference clip
                const int fi = fbase[sel];        // wave-uniform
                if (lane < 3)  outp[i * 3 + lane] = pbase[sel * 3 + lane];
                if (lane == 0) outm[i] = 1.0f;
                if (fi >= 0 && fi < npts) {
                    rfi[g] = fi;
                    const ull_t gsrc = (ull_t)(uintptr_t)(feats + (size_t)fi * D_)
                                     + (unsigned)(lane * 16);
                    const unsigned stg = stage + (unsigned)(g * 1024);
                    async_load_b128(stg, gsrc);
                    async_load_b128_o512(stg, gsrc);
                } else {                          // sentinel row -> zeros
                    const float4 z = make_float4(0.f, 0.f, 0.f, 0.f);
                    *(float4*)(dstf + lane * 4)       = z;
                    *(float4*)(dstf + lane * 4 + 128) = z;
                }
            } else {
                const float4 z = make_float4(0.f, 0.f, 0.f, 0.f);
                *(float4*)(dstf + lane * 4)       = z;
                *(float4*)(dstf + lane * 4 + 128) = z;
                if (lane < 3)  outp[i * 3 + lane] = 0.f;
                if (lane == 0) outm[i] = 0.f;
            }
        }
        // ---- all loads of this group complete, then issue async stores ----
        async_wait0();
#pragma unroll
        for (int g = 0; g < G_; ++g) {
            if (rfi[g] >= 0) {
                const int i = ib + 8 * g;
                const ull_t gdst = (ull_t)(uintptr_t)(outf + (size_t)i * D_)
                                 + (unsigned)(lane * 16);
                const unsigned stg = stage + (unsigned)(g * 1024);
                async_store_b128(gdst, stg);
                async_store_b128_o512(gdst, stg);
            }
        }
    }
    async_wait0();   // drain outstanding async stores before wave exit
}

// ===========================================================================
// Fallback: fused single-kernel version (used only if ws_size is too small)
// ===========================================================================
__global__ __launch_bounds__(256)
void n2p_fused_kernel(const int*   __restrict__ nb_mask,
                      const int*   __restrict__ seed_idx,
                      const int*   __restrict__ knn_mask,
                      const float* __restrict__ knn_pts,
                      const int*   __restrict__ knn_idx,
                      const float* __restrict__ feats,
                      float* __restrict__ out_pts,
                      float* __restrict__ out_feats,
                      float* __restrict__ out_mask,
                      int npts) {
    __shared__ int   sh_scan[256];
    __shared__ int   sh_fidx[L_];
    __shared__ float sh_pts [L_ * 3];
    __shared__ __align__(16) float sh_stage[8 * D_];

    const int p = blockIdx.x;
    const int t = threadIdx.x;
    const int s0   = t * 4;
    const int nb   = s0 >> 6;
    const int kk   = s0 & 63;
    const int node = seed_idx[p * NB_ + nb];
    const bool nbm = (nb_mask[p * NB_ + nb] != 0);

    int fidx[4]; bool vld[4]; int cnt = 0;
#pragma unroll
    for (int j = 0; j < 4; ++j) {
        const int slot = node * K_ + kk + j;
        vld[j]  = nbm && (knn_mask[slot] != 0);
        fidx[j] = knn_idx[slot];
        cnt += vld[j] ? 1 : 0;
    }
    sh_scan[t] = cnt;
    __syncthreads();
#pragma unroll
    for (int off = 1; off < 256; off <<= 1) {
        int add = 0;
        if (t >= off) add = sh_scan[t - off];
        __syncthreads();
        sh_scan[t] += add;
        __syncthreads();
    }
    int pos = sh_scan[t] - cnt;
#pragma unroll
    for (int j = 0; j < 4; ++j) {
        if (vld[j]) {
            sh_fidx[pos] = fidx[j];
            const float* pp = knn_pts + (size_t)(node * K_ + kk + j) * 3;
            sh_pts[pos * 3 + 0] = pp[0];
            sh_pts[pos * 3 + 1] = pp[1];
            sh_pts[pos * 3 + 2] = pp[2];
            ++pos;
        }
    }
    __syncthreads();
    const int n = sh_scan[255];

    const int wave = t >> 5, lane = t & 31;
    const int n_eff = (n < M_) ? n : M_;
    const bool resample = (n > M_);
    const float nf = (float)n;
    float* outp = out_pts  + (size_t)p * M_ * 3;
    float* outf = out_feats + (size_t)p * M_ * D_;
    float* outm = out_mask + (size_t)p * M_;
    const unsigned stage = lds_off_of(&sh_stage[wave * D_ + lane * 4]);

    for (int i = wave; i < M_; i += 8) {
        float* dstf = outf + (size_t)i * D_;
        if (i < n_eff) {
            int sel = resample ? (int)floorf(((float)i / 450.0f) * nf) : i;
            if (sel > L_ - 1) sel = L_ - 1;
            const int fi = sh_fidx[sel];
            if (lane < 3)  outp[i * 3 + lane] = sh_pts[sel * 3 + lane];
            if (lane == 0) outm[i] = 1.0f;
            if (fi >= 0 && fi < npts) {
                const ull_t gsrc = (ull_t)(uintptr_t)(feats + (size_t)fi * D_) + (unsigned)(lane * 16);
                const ull_t gdst = (ull_t)(uintptr_t)dstf + (unsigned)(lane * 16);
                async_wait0();
                async_load_b128(stage, gsrc);
                async_load_b128_o512(stage, gsrc);
                async_wait0();
                async_store_b128(gdst, stage);
                async_store_b128_o512(gdst, stage);
            } else {
                const float4 z = make_float4(0.f, 0.f, 0.f, 0.f);
                *(float4*)(dstf + lane * 4)       = z;
                *(float4*)(dstf + lane * 4 + 128) = z;
            }
        } else {
            const float4 z = make_float4(0.f, 0.f, 0.f, 0.f);
            *(float4*)(dstf + lane * 4)       = z;
            *(float4*)(dstf + lane * 4 + 128) = z;
            if (lane < 3)  outp[i * 3 + lane] = 0.f;
            if (lane == 0) outm[i] = 0.f;
        }
    }
    async_wait0();
}

extern "C" void kernel_launch(void* const* d_in, const int* in_sizes, int n_in,
                              void* d_out, int out_size, void* d_ws, size_t ws_size,
                              hipStream_t stream) {
    const int*   nbm   = (const int*)  d_in[0];   // ref_node_neighbor_mask  [P,16]
    const int*   sidx  = (const int*)  d_in[1];   // ref_seed_neighbor_indices [P,16]
    const int*   kmask = (const int*)  d_in[2];   // ref_node_knn_masks [NN,64]
    const float* kpts  = (const float*)d_in[3];   // ref_node_knn_points [NN,64,3]
    const int*   kidx  = (const int*)  d_in[4];   // ref_node_knn_indices [NN,64]
    const float* feats = (const float*)d_in[5];   // ref_feats_m [NPTS,256]

    const int P    = in_sizes[1] / NB_;
    const int NPTS = in_sizes[5] / D_;

    float* out       = (float*)d_out;             // concat: pts | feats | mask
    float* out_pts   = out;
    float* out_feats = out + (size_t)P * M_ * 3;
    float* out_mask  = out_feats + (size_t)P * M_ * D_;

    // workspace layout: n[P] | fidx[P][L] | pts[P][L][3]
    const size_t need = (size_t)P * 4 + (size_t)P * L_ * 4 + (size_t)P * L_ * 3 * 4;
    if (ws_size >= need) {
        int*   ws_n    = (int*)d_ws;
        int*   ws_fidx = ws_n + P;
        float* ws_pts  = (float*)(ws_fidx + (size_t)P * L_);

        n2p_compact_kernel<<<P, 256, 0, stream>>>(
            nbm, sidx, kmask, kpts, kidx, ws_n, ws_fidx, ws_pts);
        n2p_copy_kernel<<<dim3(P, SLICES_), 256, 0, stream>>>(
            ws_n, ws_fidx, ws_pts, feats, out_pts, out_feats, out_mask, NPTS);
    } else {
        n2p_fused_kernel<<<P, 256, 0, stream>>>(
            nbm, sidx, kmask, kpts, kidx, feats, out_pts, out_feats, out_mask, NPTS);
    }
}